// knn_21955872817710
// MI455X (gfx1250) — compile-verified
//
#include <hip/hip_runtime.h>

typedef __attribute__((ext_vector_type(2))) float v2f;
typedef __attribute__((ext_vector_type(8))) float v8f;

#define BATCH 4
#define NPTS  4096
#define CH    256
#define KNN   16

// ---------------------------------------------------------------------------
// Lexicographic top-16 insert (score desc, index asc) kept fully in registers.
// Matches jax.lax.top_k tie behavior (lower index wins on equal score).
// ---------------------------------------------------------------------------
__device__ __forceinline__ void topk_insert(float (&key)[16], int (&idx)[16],
                                            float s, int m) {
  if (s < key[15] || (s == key[15] && m >= idx[15])) return;
  float cs = s; int ci = m;
#pragma unroll
  for (int j = 0; j < 16; ++j) {
    bool gt = (cs > key[j]) || (cs == key[j] && ci < idx[j]);
    float ok = key[j]; int oi = idx[j];
    key[j] = gt ? cs : ok;
    idx[j] = gt ? ci : oi;
    cs = gt ? ok : cs;
    ci = gt ? oi : ci;
  }
}

// ---------------------------------------------------------------------------
// K1: inverse column norms over the POINTS axis: inv[b,c] = 1/max(||x[b,c,:]||, eps)
// ---------------------------------------------------------------------------
__global__ __launch_bounds__(256) void colnorm_kernel(const float* __restrict__ x,
                                                      float* __restrict__ inv) {
  const int bc = blockIdx.x;                       // b*CH + c
  const float* p = x + (size_t)bc * NPTS;
  float s = 0.f;
  for (int i = threadIdx.x; i < NPTS; i += 256) { float v = p[i]; s += v * v; }
  __shared__ float red[256];
  red[threadIdx.x] = s;
  __syncthreads();
  for (int off = 128; off; off >>= 1) {
    if (threadIdx.x < off) red[threadIdx.x] += red[threadIdx.x + off];
    __syncthreads();
  }
  if (threadIdx.x == 0) inv[bc] = 1.0f / fmaxf(sqrtf(red[0]), 1e-12f);
}

// ---------------------------------------------------------------------------
// K2: normalized transpose (B,C,N) -> Xn (B,N,C), LDS-tiled, coalesced both ways
// grid (CH/32, NPTS/32, B), block (32,8)
// ---------------------------------------------------------------------------
__global__ __launch_bounds__(256) void tnorm_kernel(const float* __restrict__ x,
                                                    const float* __restrict__ inv,
                                                    float* __restrict__ Xn) {
  __shared__ float tile[32][33];
  const int b = blockIdx.z, c0 = blockIdx.x * 32, n0 = blockIdx.y * 32;
  const int tx = threadIdx.x, ty = threadIdx.y;
#pragma unroll
  for (int i = 0; i < 4; ++i) {
    const int c = c0 + ty + 8 * i;
    tile[ty + 8 * i][tx] =
        x[((size_t)b * CH + c) * NPTS + n0 + tx] * inv[b * CH + c];
  }
  __syncthreads();
#pragma unroll
  for (int i = 0; i < 4; ++i) {
    const int n = n0 + ty + 8 * i;
    Xn[((size_t)b * NPTS + n) * CH + c0 + tx] = tile[tx][ty + 8 * i];
  }
}

// ---------------------------------------------------------------------------
// K3: per-point squared norms sq[b*N+n] = sum_c Xn^2 (16 lanes per row)
// grid B*NPTS/16, block 256
// ---------------------------------------------------------------------------
__global__ __launch_bounds__(256) void rowsq_kernel(const float* __restrict__ Xn,
                                                    float* __restrict__ sq) {
  const int t = threadIdx.x;
  const int row = blockIdx.x * 16 + (t >> 4);      // global b*N + n
  const float* p = Xn + (size_t)row * CH + (t & 15);
  float s = 0.f;
#pragma unroll
  for (int i = 0; i < 16; ++i) { float v = p[i * 16]; s += v * v; }
#pragma unroll
  for (int m = 8; m; m >>= 1) s += __shfl_xor(s, m, 16);
  if ((t & 15) == 0) sq[row] = s;
}

// ---------------------------------------------------------------------------
// K4: fused Gram-tile (V_WMMA_F32_16X16X4_F32) + streaming top-16.
// grid (NPTS/16, B), block 256 (8 waves). Each wave owns 32 column tiles.
// Two independent WMMA accumulation chains hide the matrix-pipe accumulate
// latency (single chain serializes on C->D dependency).
// score = 2*inner - sq[m]  (per-row constant sq[n] dropped: order-invariant)
// ---------------------------------------------------------------------------
__global__ __launch_bounds__(256) void knn_main(const float* __restrict__ Xn,
                                                const float* __restrict__ sq,
                                                int* __restrict__ out) {
  __shared__ float ldsA[16 * 260];                 // A tile, padded stride (bank-safe)
  __shared__ float scr[8][16 * 17];                // per-wave score transpose scratch
  __shared__ float mKey[16][16][16];               // [row][slot][entry]
  __shared__ int   mIdx[16][16][16];

  const int b = blockIdx.y;
  const int n0 = blockIdx.x * 16;
  const int t = threadIdx.x;
  const int wave = t >> 5, lane = t & 31;
  const int lrow = lane & 15, lhi = lane >> 4;

  // Stage 16x256 A tile (rows n0..n0+15) into LDS, coalesced.
  const float* Abase = Xn + ((size_t)b * NPTS + n0) * CH;
#pragma unroll
  for (int i = 0; i < 16; ++i) {
    const int idx = t + 256 * i;
    ldsA[(idx >> 8) * 260 + (idx & 255)] = Abase[idx];
  }
  __syncthreads();

  float key[16]; int idn[16];
#pragma unroll
  for (int j = 0; j < 16; ++j) { key[j] = -__builtin_inff(); idn[j] = 0x7FFFFFFF; }

  const int aOff = lrow * 260 + 2 * lhi;           // A frag: row=lane&15, K=2*(lane>>4)+{0,1}

  for (int tilec = wave; tilec < NPTS / 16; tilec += 8) {
    const int m0 = tilec * 16;
    // B frag (4x16, B[k][col] = Xn[m0+col][k]): col=lane&15, K=2*(lane>>4)+{0,1}
    const float* Bbase = Xn + ((size_t)b * NPTS + m0 + lrow) * CH + 2 * lhi;

    v8f acc0 = {};                                 // chain 0: k = 0,8,16,...
    v8f acc1 = {};                                 // chain 1: k = 4,12,20,...
#pragma unroll 4
    for (int k = 0; k < CH; k += 8) {
      v2f a0 = *(const v2f*)&ldsA[aOff + k];
      v2f b0 = *(const v2f*)&Bbase[k];
      v2f a1 = *(const v2f*)&ldsA[aOff + k + 4];
      v2f b1 = *(const v2f*)&Bbase[k + 4];
      acc0 = __builtin_amdgcn_wmma_f32_16x16x4_f32(
          false, a0, false, b0, (short)0, acc0, false, false);
      acc1 = __builtin_amdgcn_wmma_f32_16x16x4_f32(
          false, a1, false, b1, (short)0, acc1, false, false);
    }

    // Column of this lane is m0 + lrow for every accumulator VGPR.
    const float sqm = sq[b * NPTS + m0 + lrow];
    float* sw = &scr[wave][0];
#pragma unroll
    for (int j = 0; j < 8; ++j)                    // D row = j + 8*lhi, col = lrow
      sw[(j + 8 * lhi) * 17 + lrow] = 2.0f * (acc0[j] + acc1[j]) - sqm;

    // Re-read transposed: this lane now owns row lrow, cols 8*lhi..8*lhi+7.
    const float* rb = &sw[lrow * 17 + 8 * lhi];
    const int mb = m0 + 8 * lhi;
#pragma unroll
    for (int e = 0; e < 8; ++e) topk_insert(key, idn, rb[e], mb + e);
  }

  // Dump per-lane lists: 16 slots per row (8 waves x 2 lane-halves).
  {
    const int slot = wave * 2 + lhi;
#pragma unroll
    for (int j = 0; j < 16; ++j) { mKey[lrow][slot][j] = key[j]; mIdx[lrow][slot][j] = idn[j]; }
  }
  __syncthreads();

  // Final merge: lane r selects global top-16 of its row from 256 candidates.
  if (t < 16) {
    float fk[16]; int fi[16];
#pragma unroll
    for (int j = 0; j < 16; ++j) { fk[j] = -__builtin_inff(); fi[j] = 0x7FFFFFFF; }
    for (int slot = 0; slot < 16; ++slot) {
#pragma unroll
      for (int e = 0; e < 16; ++e)
        topk_insert(fk, fi, mKey[t][slot][e], mIdx[t][slot][e]);
    }
    const size_t base = ((size_t)b * NPTS + n0 + t) * KNN;
    int* outc = out + (size_t)BATCH * NPTS * KNN;  // second half: center indices
#pragma unroll
    for (int j = 0; j < KNN; ++j) { out[base + j] = fi[j]; outc[base + j] = n0 + t; }
  }
}

// ---------------------------------------------------------------------------
extern "C" void kernel_launch(void* const* d_in, const int* in_sizes, int n_in,
                              void* d_out, int out_size, void* d_ws, size_t ws_size,
                              hipStream_t stream) {
  const float* x = (const float*)d_in[0];          // (B, C, N, 1) fp32
  // d_in[1] = k (==16), compile-time constant here.
  int* out = (int*)d_out;                          // (2, B, N, 16) int32

  char* ws = (char*)d_ws;
  float* Xn  = (float*)ws;                                          // B*N*C f32
  float* sqv = (float*)(ws + (size_t)BATCH * NPTS * CH * 4);        // B*N f32
  float* inv = (float*)(ws + (size_t)BATCH * NPTS * CH * 4
                           + (size_t)BATCH * NPTS * 4);             // B*C f32

  colnorm_kernel<<<BATCH * CH, 256, 0, stream>>>(x, inv);

  dim3 g2(CH / 32, NPTS / 32, BATCH), b2(32, 8);
  tnorm_kernel<<<g2, b2, 0, stream>>>(x, inv, Xn);

  rowsq_kernel<<<BATCH * NPTS / 16, 256, 0, stream>>>(Xn, sqv);

  dim3 g4(NPTS / 16, BATCH);
  knn_main<<<g4, 256, 0, stream>>>(Xn, sqv, out);
}